// GCNGRU_both_22299470201222
// MI455X (gfx1250) — compile-verified
//
#include <hip/hip_runtime.h>
#include <hip/hip_bf16.h>

// ---------------- problem constants (match reference) ----------------
static constexpr int N_NODES = 50000;
static constexpr int T_SNAP  = 8;
static constexpr int E_EDGES = 800000;
static constexpr int EP      = 200000;
static constexpr int NHID    = 256;
static constexpr int NOUT    = 128;   // == H
static constexpr int H       = 128;

typedef __attribute__((ext_vector_type(2))) float v2f;
typedef __attribute__((ext_vector_type(8))) float v8f;

// ---------------- WMMA helpers: V_WMMA_F32_16X16X4_F32 ----------------
__device__ __forceinline__ v8f wmma4(v8f acc, v2f a, v2f b) {
  // 8 args: (neg_a, A, neg_b, B, c_mod, C, reuse_a, reuse_b)
  return __builtin_amdgcn_wmma_f32_16x16x4_f32(false, a, false, b, (short)0, acc,
                                               false, false);
}

__device__ __forceinline__ v8f splat8(float v) {
  v8f r = {v, v, v, v, v, v, v, v};
  return r;
}

// A fragment: 16x4 (MxK) f32, row-major source with leading dim lda.
// lane 0-15: M=lane, K pair {k0,k0+1}; lane 16-31: M=lane-16, K pair {k0+2,k0+3}
__device__ __forceinline__ v2f loadA(const float* __restrict__ A, int lda,
                                     int row0, int k0, int lane) {
  int m  = lane & 15;
  int kk = k0 + ((lane >> 4) << 1);
  const float* p = A + (size_t)(row0 + m) * lda + kk;
  v2f a; a.x = p[0]; a.y = p[1];
  return a;
}

// B fragment: 4x16 (KxN) f32 from row-major B[k][n], leading dim ldb.
__device__ __forceinline__ v2f loadB(const float* __restrict__ B, int ldb,
                                     int k0, int col0, int lane) {
  int n  = col0 + (lane & 15);
  int kk = k0 + ((lane >> 4) << 1);
  v2f b;
  b.x = B[(size_t)kk * ldb + n];
  b.y = B[(size_t)(kk + 1) * ldb + n];
  return b;
}

// ---------------- utility kernels ----------------
__global__ void k_zero(float* __restrict__ p, size_t n) {
  size_t i = (size_t)blockIdx.x * blockDim.x + threadIdx.x;
  if (i < n) p[i] = 0.0f;
}

// p[i] = bias[i & mask]
__global__ void k_fill_rows(float* __restrict__ p, const float* __restrict__ bias,
                            size_t n, int mask) {
  size_t i = (size_t)blockIdx.x * blockDim.x + threadIdx.x;
  if (i < n) p[i] = bias[i & mask];
}

// p[i] = relu(p[i] + bias[i & mask])
__global__ void k_bias_relu(float* __restrict__ p, const float* __restrict__ bias,
                            size_t n, int mask) {
  size_t i = (size_t)blockIdx.x * blockDim.x + threadIdx.x;
  if (i < n) p[i] = fmaxf(p[i] + bias[i & mask], 0.0f);
}

// COO SpMM scatter: dst[row[e], :] += val[e] * src[col[e], :]   (one wave / edge)
__global__ void k_scatter_add(const int* __restrict__ row, const int* __restrict__ col,
                              const float* __restrict__ val, const float* __restrict__ src,
                              float* __restrict__ dst, int E, int C) {
  int wave = (int)(((size_t)blockIdx.x * blockDim.x + threadIdx.x) >> 5);
  int lane = threadIdx.x & 31;
  if (wave >= E) return;
  int r = row[wave], c = col[wave];
  float v = val[wave];
  const float* s = src + (size_t)c * C;
  float* d = dst + (size_t)r * C;
  for (int j = lane; j < C; j += 32) atomicAdd(&d[j], v * s[j]);
}

// ---------------- dense GEMM: C[N,128] = A[N,256] @ B[256,128] ----------------
__global__ void k_gemm_256x128(const float* __restrict__ A, const float* __restrict__ B,
                               float* __restrict__ C, int nRows) {
  int wave = (int)(((size_t)blockIdx.x * blockDim.x + threadIdx.x) >> 5);
  int lane = threadIdx.x & 31;
  int nTiles = (nRows >> 4) * 8;            // 8 column tiles of 16
  if (wave >= nTiles) return;
  int rt = wave >> 3, ct = wave & 7;
  int row0 = rt << 4, col0 = ct << 4;

  v8f acc = splat8(0.0f);
  for (int k0 = 0; k0 < 256; k0 += 4) {
    __builtin_prefetch(A + (size_t)(row0 + (lane & 15)) * 256 + k0 + 64, 0, 1);
    v2f a = loadA(A, 256, row0, k0, lane);
    v2f b = loadB(B, 128, k0, col0, lane);
    acc = wmma4(acc, a, b);
  }
  int mHalf = (lane >> 4) << 3;
  int c = lane & 15;
  for (int r = 0; r < 8; ++r)
    C[(size_t)(row0 + mHalf + r) * 128 + col0 + c] = acc[r];
}

// ---------------- fused GRU step: h = GRU(x, h), in place ----------------
// One wave per 16-row tile; old h tile staged in a per-wave LDS slice.
// Per column-tile, the 6 transposed weight fragments (r/z/n x {W_ih,W_hh}) are
// staged ONCE per block in LDS (stride-17 padding kills bank conflicts on the
// transposing writes) and shared by all 8 waves -> 8x less global B traffic.
// Dynamic LDS: 6*128*17 (weights) + 8*16*128 (h tiles) floats = ~115 KB
// (WGP LDS is 320 KB on CDNA5).
static constexpr int GRU_WPITCH  = 17;                       // padded n-pitch
static constexpr int GRU_WSLAB   = 6 * 128 * GRU_WPITCH;     // 13056 floats
static constexpr int GRU_HSLAB   = 8 * 16 * 128;             // 16384 floats
static constexpr size_t GRU_SMEM = (size_t)(GRU_WSLAB + GRU_HSLAB) * 4;

__global__ void k_gru_step(const float* __restrict__ x, float* __restrict__ h,
                           const float* __restrict__ wih, const float* __restrict__ whh,
                           const float* __restrict__ bih, const float* __restrict__ bhh,
                           int nRows) {
  extern __shared__ float smem[];
  float* wS = smem;                 // [frag(6)][k(128)] x padded n(16), pitch 17
  float* hS = smem + GRU_WSLAB;     // 8 waves x (16 rows x 128 cols)

  int tid = threadIdx.x;
  int waveInBlk = tid >> 5;
  int lane = tid & 31;
  int tile = blockIdx.x * 8 + waveInBlk;
  bool active = (tile < (nRows >> 4));
  int row0 = tile << 4;
  float* hs = &hS[waveInBlk * 2048];

  if (active) {  // per-wave private slice: DScnt ordering within wave suffices
    for (int i = lane; i < 2048; i += 32)
      hs[i] = h[(size_t)(row0 + (i >> 7)) * 128 + (i & 127)];
  }

  int c = lane & 15;
  int nl = lane & 15;
  int mHalf = (lane >> 4) << 3;

  for (int ct = 0; ct < 8; ++ct) {
    int col0 = ct << 4;

    __syncthreads();  // previous tile's wS readers are done
    // stage 6 weight fragments, transposed: wS[(f*128+k)*17 + n] = W[(g*128+col0+n)*128 + k]
    // thread mapping: consecutive tid -> consecutive k  => coalesced global reads
    for (int i = tid; i < 6 * 2048; i += 256) {
      int k = i & 127;
      int n = (i >> 7) & 15;
      int f = i >> 11;                       // 0..5
      const float* W = (f < 3) ? wih : whh;
      int g = (f < 3) ? f : (f - 3);         // gate: 0=r, 1=z, 2=n
      wS[(f * 128 + k) * GRU_WPITCH + n] =
          W[(size_t)(g * 128 + col0 + n) * 128 + k];
    }
    __syncthreads();

    if (active) {
      v8f accR  = splat8(bih[col0 + c] + bhh[col0 + c]);          // r combined
      v8f accZ  = splat8(bih[128 + col0 + c] + bhh[128 + col0 + c]); // z combined
      v8f accXN = splat8(bih[256 + col0 + c]);                    // xn + b_in
      v8f accHN = splat8(bhh[256 + col0 + c]);                    // hn + b_hn

      for (int k0 = 0; k0 < 128; k0 += 4) {
        int kk = k0 + ((lane >> 4) << 1);
        v2f ax = loadA(x, 128, row0, k0, lane);
        v2f ah; ah.x = hs[(lane & 15) * 128 + kk];
                ah.y = hs[(lane & 15) * 128 + kk + 1];

        #define GRU_BF(f, kq) wS[((f) * 128 + (kq)) * GRU_WPITCH + nl]
        v2f bxr; bxr.x = GRU_BF(0, kk); bxr.y = GRU_BF(0, kk + 1);
        v2f bxz; bxz.x = GRU_BF(1, kk); bxz.y = GRU_BF(1, kk + 1);
        v2f bxn; bxn.x = GRU_BF(2, kk); bxn.y = GRU_BF(2, kk + 1);
        v2f bhr; bhr.x = GRU_BF(3, kk); bhr.y = GRU_BF(3, kk + 1);
        v2f bhz; bhz.x = GRU_BF(4, kk); bhz.y = GRU_BF(4, kk + 1);
        v2f bhn; bhn.x = GRU_BF(5, kk); bhn.y = GRU_BF(5, kk + 1);
        #undef GRU_BF

        accR  = wmma4(accR,  ax, bxr);
        accR  = wmma4(accR,  ah, bhr);
        accZ  = wmma4(accZ,  ax, bxz);
        accZ  = wmma4(accZ,  ah, bhz);
        accXN = wmma4(accXN, ax, bxn);
        accHN = wmma4(accHN, ah, bhn);
      }
      for (int r = 0; r < 8; ++r) {
        int m = mHalf + r;
        float rr = 1.0f / (1.0f + __expf(-accR[r]));
        float zz = 1.0f / (1.0f + __expf(-accZ[r]));
        float nn = tanhf(accXN[r] + rr * accHN[r]);
        float hold = hs[m * 128 + col0 + c];
        h[(size_t)(row0 + m) * 128 + col0 + c] = (1.0f - zz) * nn + zz * hold;
      }
    }
  }
}

// ---------------- BatchNorm (training-mode stats) ----------------
__global__ void k_bn_stats(const float* __restrict__ h, float* __restrict__ sums,
                           int N, int rowsPerBlock) {
  int c = threadIdx.x;                    // 128 threads = 128 cols
  int r0 = blockIdx.x * rowsPerBlock;
  int r1 = r0 + rowsPerBlock; if (r1 > N) r1 = N;
  float s = 0.0f, s2 = 0.0f;
  for (int r = r0; r < r1; ++r) {
    float v = h[(size_t)r * 128 + c];
    s += v; s2 += v * v;
  }
  atomicAdd(&sums[c], s);
  atomicAdd(&sums[128 + c], s2);
}

__global__ void k_bn_apply(float* __restrict__ h, const float* __restrict__ sums,
                           const float* __restrict__ gamma, const float* __restrict__ beta,
                           int N) {
  size_t i = (size_t)blockIdx.x * blockDim.x + threadIdx.x;
  if (i >= (size_t)N * 128) return;
  int c = (int)(i & 127);
  float invN = 1.0f / (float)N;
  float mean = sums[c] * invN;
  float var  = sums[128 + c] * invN - mean * mean;
  h[i] = (h[i] - mean) * rsqrtf(var + 1e-5f) * gamma[c] + beta[c];
}

// ---------------- fused 2-layer MLP + log_softmax (edge & node predictors) ----
// input row r = concat(featA[idxA ? idxA[r] : r], featB[idxB ? idxB[r] : r]) (256)
// layer1: [256]->[128] WMMA + relu; per-N-tile w1 slab staged once per block in
// LDS ([k][n] layout, conflict-free) and shared by all 8 waves; hidden tile in
// per-wave LDS. layer2: [128]->[2] VALU dot + stable log_softmax (shfl_xor 16).
// Dynamic LDS: 16*256 (w1 slab) + 8*16*128 (hidden) floats = 80 KB.
static constexpr int MLP_WSLAB   = 16 * 256;                 // 4096 floats
static constexpr int MLP_HSLAB   = 8 * 16 * 128;             // 16384 floats
static constexpr size_t MLP_SMEM = (size_t)(MLP_WSLAB + MLP_HSLAB) * 4;

__global__ void k_pair_mlp(const float* __restrict__ featA, const float* __restrict__ featB,
                           const int* __restrict__ idxA, const int* __restrict__ idxB,
                           const float* __restrict__ w1, const float* __restrict__ b1,
                           const float* __restrict__ w2, const float* __restrict__ b2,
                           float* __restrict__ out, int nRows) {
  extern __shared__ float smem[];
  float* wSlab  = smem;               // [k(256)][n(16)]
  float* hidAll = smem + MLP_WSLAB;   // 8 waves x (16 rows x 128)

  int tid = threadIdx.x;
  int waveInBlk = tid >> 5;
  int lane = tid & 31;
  int tile = blockIdx.x * 8 + waveInBlk;
  bool active = (tile < (nRows >> 4));
  int e0 = tile << 4;
  float* hid = &hidAll[waveInBlk * 2048];

  int m = lane & 15;
  int row = active ? (e0 + m) : 0;
  int ia = idxA ? idxA[row] : row;
  int ib = idxB ? idxB[row] : row;
  const float* rowA = featA + (size_t)ia * 128;
  const float* rowB = featB + (size_t)ib * 128;

  int c = lane & 15;
  int nl = lane & 15;
  int mHalf = (lane >> 4) << 3;

  for (int nt = 0; nt < 8; ++nt) {
    int col0 = nt << 4;

    __syncthreads();
    // stage w1 columns [col0, col0+16): wSlab[k*16 + n] = w1[k*128 + col0 + n]
    for (int i = tid; i < MLP_WSLAB; i += 256) {
      int n = i & 15;
      int k = i >> 4;
      wSlab[i] = w1[(size_t)k * 128 + col0 + n];
    }
    __syncthreads();

    if (active) {
      v8f acc = splat8(b1[col0 + c]);
      for (int k0 = 0; k0 < 256; k0 += 4) {
        int kk = k0 + ((lane >> 4) << 1);
        v2f a;
        if (kk < 128) { a.x = rowA[kk];       a.y = rowA[kk + 1]; }
        else          { a.x = rowB[kk - 128]; a.y = rowB[kk - 127]; }
        v2f b; b.x = wSlab[kk * 16 + nl]; b.y = wSlab[(kk + 1) * 16 + nl];
        acc = wmma4(acc, a, b);
      }
      for (int r = 0; r < 8; ++r)
        hid[(mHalf + r) * 128 + col0 + c] = fmaxf(acc[r], 0.0f);
    }
  }

  if (!active) return;  // no barriers past this point

  // layer 2: 32 lanes = 16 rows x 2 classes
  int cls = lane >> 4;
  const float* hrow = &hid[m * 128];
  float z = b2[cls];
  for (int j = 0; j < 128; ++j) z += hrow[j] * w2[j * 2 + cls];
  float zo = __shfl_xor(z, 16, 32);
  float mx = fmaxf(z, zo);
  float lse = mx + logf(__expf(z - mx) + __expf(zo - mx));
  out[(size_t)(e0 + m) * 2 + cls] = z - lse;
}

// ---------------- node-predictor graph prep ----------------
__global__ void k_deg(const int* __restrict__ row, const int* __restrict__ col,
                      float* __restrict__ deg, int E) {
  int i = blockIdx.x * blockDim.x + threadIdx.x;
  if (i >= E) return;
  int r = row[i];
  if (r != col[i]) atomicAdd(&deg[r], 1.0f);
}

// attention-weighted neighbor aggregation (one wave / edge)
__global__ void k_attn_neigh(const float* __restrict__ emb, const int* __restrict__ row,
                             const int* __restrict__ col, const float* __restrict__ attw,
                             const float* __restrict__ attb, const float* __restrict__ deg,
                             float* __restrict__ neigh, int E) {
  int wave = (int)(((size_t)blockIdx.x * blockDim.x + threadIdx.x) >> 5);
  int lane = threadIdx.x & 31;
  if (wave >= E) return;
  int r = row[wave], cc = col[wave];
  if (r == cc) return;                       // mask == 0
  const float* er = emb + (size_t)r  * 128;
  const float* ec = emb + (size_t)cc * 128;
  float p = 0.0f;
  for (int j = lane; j < 128; j += 32)
    p += er[j] * attw[j] + ec[j] * attw[128 + j];
  for (int off = 16; off > 0; off >>= 1) p += __shfl_xor(p, off, 32);
  float w  = 1.0f / (1.0f + __expf(-(p + attb[0])));
  float ew = w / deg[r];                     // deg[r] >= 1 for any masked edge
  for (int j = lane; j < 128; j += 32)
    atomicAdd(&neigh[(size_t)r * 128 + j], ew * ec[j]);
}

// ---------------- host driver ----------------
static inline int nblk(size_t n, int t) { return (int)((n + t - 1) / t); }

extern "C" void kernel_launch(void* const* d_in, const int* in_sizes, int n_in,
                              void* d_out, int out_size, void* d_ws, size_t ws_size,
                              hipStream_t stream) {
  const int*   adj_idx = (const int*)d_in[0];     // [T+1][2][E]
  const float* adj_val = (const float*)d_in[1];   // [T+1][E]
  const int*   edges   = (const int*)d_in[2];     // [2][EP]
  const float* gcn_w1  = (const float*)d_in[3];
  const float* gcn_b1  = (const float*)d_in[4];
  const float* gcn_w2  = (const float*)d_in[5];
  const float* gcn_b2  = (const float*)d_in[6];
  const float* gru_wih = (const float*)d_in[7];
  const float* gru_whh = (const float*)d_in[8];
  const float* gru_bih = (const float*)d_in[9];
  const float* gru_bhh = (const float*)d_in[10];
  const float* bn_g    = (const float*)d_in[11];
  const float* bn_b    = (const float*)d_in[12];
  const float* attn_w  = (const float*)d_in[13];
  const float* attn_b  = (const float*)d_in[14];
  const float* node_w1 = (const float*)d_in[15];
  const float* node_b1 = (const float*)d_in[16];
  const float* node_w2 = (const float*)d_in[17];
  const float* node_b2 = (const float*)d_in[18];
  const float* edge_w1 = (const float*)d_in[19];
  const float* edge_b1 = (const float*)d_in[20];
  const float* edge_w2 = (const float*)d_in[21];
  const float* edge_b2 = (const float*)d_in[22];

  // workspace layout (floats)
  float* ws    = (float*)d_ws;
  float* x1    = ws;                                   // N*256
  float* xw    = x1    + (size_t)N_NODES * NHID;       // N*128
  float* x2    = xw    + (size_t)N_NODES * NOUT;       // N*128
  float* h     = x2    + (size_t)N_NODES * NOUT;       // N*128
  float* neigh = h     + (size_t)N_NODES * H;          // N*128
  float* deg   = neigh + (size_t)N_NODES * H;          // N
  float* sums  = deg   + N_NODES;                      // 256

  const size_t n1 = (size_t)N_NODES * NHID;            // 12.8M
  const size_t n2 = (size_t)N_NODES * NOUT;            // 6.4M
  const int edgeWaves = nblk((size_t)E_EDGES * 32, 256);

  // h0 = 0
  k_zero<<<nblk(n2, 256), 256, 0, stream>>>(h, n2);

  for (int t = 0; t < T_SNAP; ++t) {
    const int*   row = adj_idx + (size_t)t * 2 * E_EDGES;
    const int*   col = row + E_EDGES;
    const float* val = adj_val + (size_t)t * E_EDGES;

    // layer 1: x1 = relu(spmm(A, W1) + b1)
    k_zero<<<nblk(n1, 256), 256, 0, stream>>>(x1, n1);
    k_scatter_add<<<edgeWaves, 256, 0, stream>>>(row, col, val, gcn_w1, x1,
                                                 E_EDGES, NHID);
    k_bias_relu<<<nblk(n1, 256), 256, 0, stream>>>(x1, gcn_b1, n1, NHID - 1);

    // xw = x1 @ W2  (WMMA)
    k_gemm_256x128<<<(N_NODES / 16) * 8 / 8, 256, 0, stream>>>(x1, gcn_w2, xw,
                                                               N_NODES);

    // layer 2: x2 = spmm(A, xw) + b2
    k_fill_rows<<<nblk(n2, 256), 256, 0, stream>>>(x2, gcn_b2, n2, NOUT - 1);
    k_scatter_add<<<edgeWaves, 256, 0, stream>>>(row, col, val, xw, x2,
                                                 E_EDGES, NOUT);

    // GRU step (fused WMMA, block-shared weight slabs in LDS, h in place)
    k_gru_step<<<nblk(N_NODES / 16, 8), 256, GRU_SMEM, stream>>>(
        x2, h, gru_wih, gru_whh, gru_bih, gru_bhh, N_NODES);
  }

  // BatchNorm1d (batch statistics)
  k_zero<<<1, 256, 0, stream>>>(sums, 256);
  k_bn_stats<<<nblk(N_NODES, 256), 128, 0, stream>>>(h, sums, N_NODES, 256);
  k_bn_apply<<<nblk(n2, 256), 256, 0, stream>>>(h, sums, bn_g, bn_b, N_NODES);

  // ---- edge predictor ----
  float* outEdges = (float*)d_out;                     // [EP,2]
  float* outNodes = outEdges + (size_t)EP * 2;         // [N,2]
  k_pair_mlp<<<nblk(EP / 16, 8), 256, MLP_SMEM, stream>>>(
      h, h, edges, edges + EP, edge_w1, edge_b1, edge_w2, edge_b2, outEdges, EP);

  // ---- node predictor on snapshot T ----
  const int* rowT = adj_idx + (size_t)T_SNAP * 2 * E_EDGES;
  const int* colT = rowT + E_EDGES;
  k_zero<<<nblk((size_t)N_NODES, 256), 256, 0, stream>>>(deg, N_NODES);
  k_deg<<<nblk(E_EDGES, 256), 256, 0, stream>>>(rowT, colT, deg, E_EDGES);
  k_zero<<<nblk(n2, 256), 256, 0, stream>>>(neigh, n2);
  k_attn_neigh<<<edgeWaves, 256, 0, stream>>>(h, rowT, colT, attn_w, attn_b, deg,
                                              neigh, E_EDGES);
  k_pair_mlp<<<nblk(N_NODES / 16, 8), 256, MLP_SMEM, stream>>>(
      h, neigh, nullptr, nullptr, node_w1, node_b1, node_w2, node_b2, outNodes,
      N_NODES);
}